// MMDLoss_9122510537222
// MI455X (gfx1250) — compile-verified
//
#include <hip/hip_runtime.h>

typedef float v2f __attribute__((ext_vector_type(2)));
typedef float v8f __attribute__((ext_vector_type(8)));
typedef int   v4i __attribute__((ext_vector_type(4)));

#define N_TOT 4096
#define D_DIM 512
#define HALF  2048

// workspace layout (floats)
#define SQ_OFF    0      // 4096 row squared norms
#define CSS_OFF   4096   // 512 colsum(src)
#define CST_OFF   4608   // 512 colsum(tgt)
#define ACC_OFF   5120   // [0]=sumsq_src [1]=sumsq_tgt [2]=result
#define CONST_OFF 5136   // 15 floats: c_xx[5], c_yy[5], c_xy[5]  (log2e/bw/2^s)

#define BT  128   // block output tile (128x128)
#define KC  16    // K chunk staged in LDS (double buffered)
#define NCH (D_DIM / KC)
#define LDA 18    // padded LDS row stride: even (8B-aligned b64) + conflict-free

// ---- gfx1250 async global->LDS path (ASYNCcnt), with safe fallback --------
#if defined(__has_builtin)
# if __has_builtin(__builtin_amdgcn_global_load_async_to_lds_b128)
#  define USE_ASYNC_LDS 1
# endif
#endif
#ifndef USE_ASYNC_LDS
# define USE_ASYNC_LDS 0
#endif

#if USE_ASYNC_LDS
// signature: (int4 addrspace(1)* src, int4 addrspace(3)* dst, imm offset, imm cpol)
# define ASYNC_CP16(gsrc, ldst)                                        \
    __builtin_amdgcn_global_load_async_to_lds_b128(                    \
        (__attribute__((address_space(1))) v4i*)(gsrc),                \
        (__attribute__((address_space(3))) v4i*)(ldst), 0, 0)
# if __has_builtin(__builtin_amdgcn_s_wait_asynccnt)
#  define WAIT_ASYNC() __builtin_amdgcn_s_wait_asynccnt(0)
# else
#  define WAIT_ASYNC() asm volatile("s_wait_asynccnt 0x0" ::: "memory")
# endif
#else
# define WAIT_ASYNC() ((void)0)
#endif

__global__ void k_zero(float* ws) {
    if (threadIdx.x < 3) ws[ACC_OFF + threadIdx.x] = 0.0f;
}

// one wave32 per row: squared norm + domain-wise sum of squared norms
__global__ void k_rowsq(const float* __restrict__ x, float* __restrict__ ws) {
    const int wave = threadIdx.x >> 5;
    const int lane = threadIdx.x & 31;
    const int row  = blockIdx.x * 8 + wave;
    const float4* p = reinterpret_cast<const float4*>(x + (size_t)row * D_DIM);
    float s = 0.f;
    for (int c = lane; c < D_DIM / 4; c += 32) {
        float4 v = p[c];
        s += v.x * v.x + v.y * v.y + v.z * v.z + v.w * v.w;
    }
    for (int off = 16; off; off >>= 1) s += __shfl_xor(s, off, 32);
    if (lane == 0) {
        ws[SQ_OFF + row] = s;
        atomicAdd(&ws[ACC_OFF + (row < HALF ? 0 : 1)], s);
    }
}

// column sums per domain: 512 threads total
__global__ void k_colsum(const float* __restrict__ x, float* __restrict__ ws) {
    const int c = blockIdx.x * blockDim.x + threadIdx.x;
    if (c >= D_DIM) return;
    float s = 0.f, t = 0.f;
    for (int r = 0; r < HALF; ++r)      s += x[(size_t)r * D_DIM + c];
    for (int r = HALF; r < N_TOT; ++r)  t += x[(size_t)r * D_DIM + c];
    ws[CSS_OFF + c] = s;
    ws[CST_OFF + c] = t;
}

// bandwidths -> 15 exp2 constants (log2e folded in).
// Uses  sum(G block) = <colsum_a, colsum_b>.
__global__ void k_consts(float* ws) {
    __shared__ float r0[512], r1[512], r2[512];
    const int c = threadIdx.x;
    const float sv = ws[CSS_OFF + c], tv = ws[CST_OFF + c];
    r0[c] = sv * sv; r1[c] = tv * tv; r2[c] = sv * tv;
    __syncthreads();
    for (int s = 256; s > 0; s >>= 1) {
        if (c < s) { r0[c] += r0[c + s]; r1[c] += r1[c + s]; r2[c] += r2[c + s]; }
        __syncthreads();
    }
    if (c == 0) {
        const double Gss = r0[0], Gtt = r1[0], Gst = r2[0];
        const double sqs = ws[ACC_OFF + 0], sqt = ws[ACC_OFF + 1];
        const double m = (double)N_TOT;
        const double denom = m * m - m;
        const double sd_xx = 4.0 * m * sqs - 8.0 * Gss;
        const double sd_yy = 4.0 * m * sqt - 8.0 * Gtt;
        const double sd_xy = 2.0 * m * (sqs + sqt) - 2.0 * (Gss + Gtt + 2.0 * Gst);
        const double bxx = sd_xx / denom * 0.25;   // / kernel_mul^(num//2) = /4
        const double byy = sd_yy / denom * 0.25;
        const double bxy = sd_xy / denom * 0.25;
        const double L2E = 1.4426950408889634;     // exp(-x) = exp2(-x*L2E)
        double p = 1.0;
        for (int i = 0; i < 5; ++i) {
            ws[CONST_OFF + i]      = (float)(L2E / (bxx * p));
            ws[CONST_OFF + 5 + i]  = (float)(L2E / (byy * p));
            ws[CONST_OFF + 10 + i] = (float)(L2E / (bxy * p));
            p *= 2.0;
        }
    }
}

// stage one 128xKC panel pair into LDS (async on gfx1250, sync fallback)
__device__ __forceinline__ void stage_chunk(const float* __restrict__ x,
                                            int i0, int j0, int k0,
                                            float* As, float* Bs, int tid) {
    const int r_in = tid >> 2;          // 0..63
    const int c4   = (tid & 3) << 2;    // 0,4,8,12
#pragma unroll
    for (int p = 0; p < 2; ++p) {
        const int r = r_in + p * 64;
        const float* ga = x + (size_t)(i0 + r) * D_DIM + k0 + c4;
        const float* gb = x + (size_t)(j0 + r) * D_DIM + k0 + c4;
        float* la = &As[r * LDA + c4];
        float* lb = &Bs[r * LDA + c4];
#if USE_ASYNC_LDS
        ASYNC_CP16(ga, la);
        ASYNC_CP16(gb, lb);
#else
        *reinterpret_cast<float4*>(la) = *reinterpret_cast<const float4*>(ga);
        *reinterpret_cast<float4*>(lb) = *reinterpret_cast<const float4*>(gb);
#endif
    }
}

// fused Gram (WMMA f32 16x16x4) + Gaussian-kernel epilogue + global reduce
__global__ __launch_bounds__(256, 4) void k_mmd(const float* __restrict__ x,
                                                const float* __restrict__ ws,
                                                float* __restrict__ res) {
    __shared__ float As[2][BT * LDA];
    __shared__ float Bs[2][BT * LDA];
    __shared__ float sqI[BT], sqJ[BT];
    __shared__ float wpart[8];

    const int bi = blockIdx.x, bj = blockIdx.y;
    const int i0 = bi * BT, j0 = bj * BT;
    const int tid  = threadIdx.x;
    const int lane = tid & 31, wave = tid >> 5;
    const int wr = wave >> 1;      // 0..3 -> 32-row strip
    const int wc = wave & 1;       // 0..1 -> 64-col strip

    if (tid < BT) sqI[tid] = ws[SQ_OFF + i0 + tid];
    else          sqJ[tid - BT] = ws[SQ_OFF + j0 + (tid - BT)];

    v8f acc[2][4] = {};

    const int lrow = lane & 15;          // M (A frag) / N (B frag)
    const int lk   = (lane >> 4) << 1;   // K half offset: 0 or 2

    // prologue: stage chunk 0
    stage_chunk(x, i0, j0, 0, As[0], Bs[0], tid);
    WAIT_ASYNC();
    __syncthreads();

    for (int c = 0; c < NCH; ++c) {
        const int cur = c & 1;
        if (c + 1 < NCH)   // prefetch next panel into alternate buffer
            stage_chunk(x, i0, j0, (c + 1) * KC, As[cur ^ 1], Bs[cur ^ 1], tid);

        const float* Ac = As[cur];
        const float* Bc = Bs[cur];
#pragma unroll
        for (int kk = 0; kk < KC; kk += 4) {
            v2f a[2], b[4];
#pragma unroll
            for (int mi = 0; mi < 2; ++mi)
                a[mi] = *reinterpret_cast<const v2f*>(
                    &Ac[(wr * 32 + mi * 16 + lrow) * LDA + kk + lk]);
#pragma unroll
            for (int ni = 0; ni < 4; ++ni)
                b[ni] = *reinterpret_cast<const v2f*>(
                    &Bc[(wc * 64 + ni * 16 + lrow) * LDA + kk + lk]);
#pragma unroll
            for (int mi = 0; mi < 2; ++mi)
#pragma unroll
                for (int ni = 0; ni < 4; ++ni)
                    acc[mi][ni] = __builtin_amdgcn_wmma_f32_16x16x4_f32(
                        false, a[mi], false, b[ni], (short)0, acc[mi][ni],
                        false, false);
        }
        WAIT_ASYNC();      // my async writes for chunk c+1 have landed
        __syncthreads();   // everyone done reading cur + everyone's writes visible
    }

    // ---- epilogue (constants loaded only now: keep matmul-loop pressure low)
    const bool same = (bi < 16) == (bj < 16);   // domain boundary at tile 16
    const int  locoff = (bi < 16) ? 0 : 5;
    float invxy[5], invloc[5];
#pragma unroll
    for (int s = 0; s < 5; ++s) {
        invxy[s]  = ws[CONST_OFF + 10 + s];
        invloc[s] = ws[CONST_OFF + locoff + s];
    }
    const float wLOC = same ? (1.0f / (2048.0f * 2048.0f)) : 0.0f;
    const float wXY  = -2.0f / (4096.0f * 4096.0f);
    const int   mbase = (lane >> 4) << 3;   // C layout: M = vgpr + 8*(lane>=16)

    float part = 0.f;
    // fully unrolled: acc indices must stay compile-time constants so the
    // accumulator array is never demoted to scratch
#pragma unroll
    for (int mi = 0; mi < 2; ++mi) {
#pragma unroll
        for (int ni = 0; ni < 4; ++ni) {
            const v8f cv = acc[mi][ni];
#pragma unroll
            for (int v = 0; v < 8; ++v) {
                const int il = wr * 32 + mi * 16 + mbase + v;
                const int jl = wc * 64 + ni * 16 + lrow;
                const float d2 = sqI[il] + sqJ[jl] - 2.0f * cv[v];
                float kxy = 0.f;
#pragma unroll
                for (int s = 0; s < 5; ++s) kxy += exp2f(-d2 * invxy[s]);
                part += wXY * kxy;
                if (same) {
                    float kl = 0.f;
#pragma unroll
                    for (int s = 0; s < 5; ++s) kl += exp2f(-d2 * invloc[s]);
                    part += wLOC * kl;
                }
            }
        }
    }

    for (int off = 16; off; off >>= 1) part += __shfl_xor(part, off, 32);
    if (lane == 0) wpart[wave] = part;
    __syncthreads();
    if (wave == 0) {
        float v = (lane < 8) ? wpart[lane] : 0.f;
        for (int off = 4; off; off >>= 1) v += __shfl_xor(v, off, 32);
        if (lane == 0) atomicAdd(res, v);
    }
}

__global__ void k_final(const float* __restrict__ ws, float* __restrict__ out) {
    if (threadIdx.x == 0) out[0] = ws[ACC_OFF + 2];
}

extern "C" void kernel_launch(void* const* d_in, const int* in_sizes, int n_in,
                              void* d_out, int out_size, void* d_ws, size_t ws_size,
                              hipStream_t stream) {
    (void)in_sizes; (void)n_in; (void)out_size; (void)ws_size;
    const float* x = (const float*)d_in[0];   // latent 4096x512 f32 (domain unused)
    float* ws  = (float*)d_ws;
    float* out = (float*)d_out;

    k_zero  <<<1, 64, 0, stream>>>(ws);
    k_rowsq <<<N_TOT / 8, 256, 0, stream>>>(x, ws);
    k_colsum<<<2, 256, 0, stream>>>(x, ws);
    k_consts<<<1, 512, 0, stream>>>(ws);
    dim3 grid(N_TOT / BT, N_TOT / BT);
    k_mmd   <<<grid, 256, 0, stream>>>(x, ws, ws + ACC_OFF + 2);
    k_final <<<1, 32, 0, stream>>>(ws, out);
}